// GATBlock_15092515078187
// MI455X (gfx1250) — compile-verified
//
#include <hip/hip_runtime.h>
#include <math.h>

#define NNODES 100000
#define NEDGES 1000000
#define EPEDGE (NEDGES + NNODES)   // edges + self loops = 1,100,000
#define DIM 64
#define NH 4
#define NG 64
#define EPSN 1e-5f
#define LRELU_SLOPE 0.2f

typedef float v2f __attribute__((ext_vector_type(2)));
typedef float v8f __attribute__((ext_vector_type(8)));

// ------------------------------------------------------------------
// Kernel A: xl = x @ Wl, xr = x @ Wr  (fp32 WMMA 16x16x4)
// Block = 256 threads = 8 waves; block owns 32 rows of x (2 M-tiles/wave).
// Waves 0-3 -> Wl column tiles n0=0,16,32,48 ; waves 4-7 -> Wr tiles.
// B fragments (4x16 f32 per K-step) are loaded ONCE per wave into
// registers (16 frags = 32 VGPRs) and reused across both M-tiles.
// A frag (16x4 f32): lanes 0-15 hold K={k0,k0+1}, lanes 16-31 K={k0+2,k0+3}
// B frag (4x16 f32): VGPR0 = rows k0 (lo-lanes) / k0+2 (hi-lanes), VGPR1 = +1
// ------------------------------------------------------------------
__global__ __launch_bounds__(256) void proj_wmma(const float* __restrict__ x,
                                                 const float* __restrict__ Wl,
                                                 const float* __restrict__ Wr,
                                                 float* __restrict__ xl,
                                                 float* __restrict__ xr) {
  const int wave = threadIdx.x >> 5;
  const int lane = threadIdx.x & 31;
  const float* __restrict__ W   = (wave < 4) ? Wl : Wr;
  float*       __restrict__ out = (wave < 4) ? xl : xr;
  const int n0    = (wave & 3) << 4;
  const int l15   = lane & 15;
  const int hi    = lane >> 4;               // 0 | 1
  const int khalf = hi << 1;                 // 0 | 2
  const int col   = n0 + l15;

  // ---- preload all 16 B fragments for this column tile (K = 0..63) ----
  v2f b[16];
#pragma unroll
  for (int k = 0; k < 16; ++k) {
    b[k].x = W[(size_t)(4 * k + khalf)     * DIM + col];
    b[k].y = W[(size_t)(4 * k + khalf + 1) * DIM + col];
  }

  const int m_base = blockIdx.x << 5;        // 32 rows per block, N % 32 == 0
#pragma unroll
  for (int t = 0; t < 2; ++t) {
    const int m0 = m_base + (t << 4);
    const size_t row = (size_t)(m0 + l15);
    v8f c = {0.f, 0.f, 0.f, 0.f, 0.f, 0.f, 0.f, 0.f};
#pragma unroll
    for (int k = 0; k < 16; ++k) {
      v2f a = *(const v2f*)(x + row * DIM + (4 * k + khalf));   // 8B aligned
      c = __builtin_amdgcn_wmma_f32_16x16x4_f32(false, a, false, b[k],
                                                (short)0, c, false, false);
    }
    // D layout: VGPR i -> M = m0 + i + 8*hi, N = col
    const int rbase = m0 + (hi << 3);
#pragma unroll
    for (int i = 0; i < 8; ++i)
      out[(size_t)(rbase + i) * DIM + col] = c[i];
  }
}

// ------------------------------------------------------------------
// Workspace init (accumulators must be clean every launch)
// ------------------------------------------------------------------
__global__ void init_ws(float* __restrict__ agg, float* __restrict__ mmax,
                        float* __restrict__ denom, float* __restrict__ gsum,
                        float* __restrict__ vsum, float* __restrict__ cnt) {
  int id = blockIdx.x * blockDim.x + threadIdx.x;
  if (id < NNODES * DIM) agg[id] = 0.f;
  if (id < NNODES * NH) { mmax[id] = -INFINITY; denom[id] = 0.f; }
  if (id < NG * DIM)    { gsum[id] = 0.f; vsum[id] = 0.f; }
  if (id < NG)          cnt[id] = 0.f;
}

// ------------------------------------------------------------------
// Kernel B: per-edge attention logits + segment max.
// One wave / edge; lane owns channels {2*lane, 2*lane+1} (head = lane>>3).
// ------------------------------------------------------------------
__global__ __launch_bounds__(256) void edge_score(const float* __restrict__ xl,
                                                  const float* __restrict__ xr,
                                                  const int*   __restrict__ ei,
                                                  const float* __restrict__ att,
                                                  float* __restrict__ score,
                                                  float* __restrict__ mmax) {
  const int lane = threadIdx.x & 31;
  const int e = (blockIdx.x * blockDim.x + threadIdx.x) >> 5;
  if (e >= EPEDGE) return;
  int src, dst;
  if (e < NEDGES) { src = ei[e]; dst = ei[NEDGES + e]; }
  else            { src = e - NEDGES; dst = src; }

  const int c0 = lane << 1;
  v2f a = *(const v2f*)(xl + (size_t)src * DIM + c0);
  v2f b = *(const v2f*)(xr + (size_t)dst * DIM + c0);
  float e0 = a.x + b.x; e0 = (e0 >= 0.f) ? e0 : LRELU_SLOPE * e0;
  float e1 = a.y + b.y; e1 = (e1 >= 0.f) ? e1 : LRELU_SLOPE * e1;
  float s = e0 * att[c0] + e1 * att[c0 + 1];
  // reduce the 8 lanes belonging to one head
  s += __shfl_xor(s, 1, 32);
  s += __shfl_xor(s, 2, 32);
  s += __shfl_xor(s, 4, 32);
  if ((lane & 7) == 0) {
    const int h = lane >> 3;
    score[(size_t)e * NH + h] = s;
    atomicMax(&mmax[(size_t)dst * NH + h], s);   // GLOBAL_ATOMIC_MAX_NUM_F32
  }
}

// ------------------------------------------------------------------
// Kernel C: p = exp(score - max[dst]);  denom[dst] += p
// ------------------------------------------------------------------
__global__ void exp_denom(const int* __restrict__ ei,
                          const float* __restrict__ mmax,
                          float* __restrict__ score,
                          float* __restrict__ denom) {
  int id = blockIdx.x * blockDim.x + threadIdx.x;
  if (id >= EPEDGE * NH) return;
  int e = id >> 2, h = id & 3;
  int dst = (e < NEDGES) ? ei[NEDGES + e] : (e - NEDGES);
  float p = __expf(score[id] - mmax[(size_t)dst * NH + h]);
  score[id] = p;
  atomicAdd(&denom[(size_t)dst * NH + h], p);
}

// ------------------------------------------------------------------
// Kernel D: agg[dst] += (p/denom[dst]) * xl[src]   (wave per edge)
// ------------------------------------------------------------------
__global__ __launch_bounds__(256) void aggregate(const float* __restrict__ xl,
                                                 const int*   __restrict__ ei,
                                                 const float* __restrict__ score,
                                                 const float* __restrict__ denom,
                                                 float* __restrict__ agg) {
  const int lane = threadIdx.x & 31;
  const int e = (blockIdx.x * blockDim.x + threadIdx.x) >> 5;
  if (e >= EPEDGE) return;
  int src, dst;
  if (e < NEDGES) { src = ei[e]; dst = ei[NEDGES + e]; }
  else            { src = e - NEDGES; dst = src; }
  const int c0 = lane << 1;
  const int h  = lane >> 3;
  float alpha = score[(size_t)e * NH + h] / denom[(size_t)dst * NH + h];
  v2f a = *(const v2f*)(xl + (size_t)src * DIM + c0);
  atomicAdd(&agg[(size_t)dst * DIM + c0],     alpha * a.x);
  atomicAdd(&agg[(size_t)dst * DIM + c0 + 1], alpha * a.y);
}

// ------------------------------------------------------------------
// Kernel E: h = ELU(agg + bias + x); per-graph channel sums + counts
// ------------------------------------------------------------------
__global__ void post_elu(const float* __restrict__ x,
                         const float* __restrict__ bias,
                         const int*   __restrict__ batch,
                         float* __restrict__ h,           // in: agg, out: h
                         float* __restrict__ gsum,
                         float* __restrict__ cnt) {
  int id = blockIdx.x * blockDim.x + threadIdx.x;
  if (id >= NNODES * DIM) return;
  int n = id >> 6, c = id & 63;
  float v = h[id] + bias[c] + x[id];
  v = (v > 0.f) ? v : expm1f(v);
  h[id] = v;
  int g = batch[n];
  atomicAdd(&gsum[g * DIM + c], v);
  if (c == 0) atomicAdd(&cnt[g], 1.f);
}

// ------------------------------------------------------------------
// Kernel F1: center by scaled mean, accumulate variance sums
// ------------------------------------------------------------------
__global__ void center_var(const int* __restrict__ batch,
                           const float* __restrict__ mean_scale,
                           const float* __restrict__ gsum,
                           const float* __restrict__ cnt,
                           float* __restrict__ h,          // in/out: centered
                           float* __restrict__ vsum) {
  int id = blockIdx.x * blockDim.x + threadIdx.x;
  if (id >= NNODES * DIM) return;
  int n = id >> 6, c = id & 63;
  int g = batch[n];
  float cv = fmaxf(cnt[g], 1.f);
  float mean = gsum[g * DIM + c] / cv;
  float v = h[id] - mean_scale[c] * mean;
  h[id] = v;
  atomicAdd(&vsum[g * DIM + c], v * v);
}

// ------------------------------------------------------------------
// Kernel F2: out = weight * h / sqrt(var + eps) + bias
// ------------------------------------------------------------------
__global__ void finalize(const int* __restrict__ batch,
                         const float* __restrict__ weight,
                         const float* __restrict__ gbias,
                         const float* __restrict__ cnt,
                         const float* __restrict__ vsum,
                         const float* __restrict__ h,
                         float* __restrict__ out) {
  int id = blockIdx.x * blockDim.x + threadIdx.x;
  if (id >= NNODES * DIM) return;
  int n = id >> 6, c = id & 63;
  int g = batch[n];
  float var = vsum[g * DIM + c] / fmaxf(cnt[g], 1.f);
  out[id] = weight[c] * h[id] * rsqrtf(var + EPSN) + gbias[c];
}

// ------------------------------------------------------------------
extern "C" void kernel_launch(void* const* d_in, const int* in_sizes, int n_in,
                              void* d_out, int out_size, void* d_ws, size_t ws_size,
                              hipStream_t stream) {
  const float* x          = (const float*)d_in[0];
  const int*   ei         = (const int*)  d_in[1];   // [2,E] flat: src | dst
  const int*   batch      = (const int*)  d_in[2];
  const float* Wl         = (const float*)d_in[3];
  const float* Wr         = (const float*)d_in[4];
  const float* att        = (const float*)d_in[5];   // [H,16] flat == [64]
  const float* bias_gat   = (const float*)d_in[6];
  const float* gn_weight  = (const float*)d_in[7];
  const float* gn_bias    = (const float*)d_in[8];
  const float* gn_mscale  = (const float*)d_in[9];
  float* out = (float*)d_out;

  // workspace layout (floats)
  float* ws    = (float*)d_ws;
  float* xl    = ws;                         // N*64
  float* xr    = xl    + (size_t)NNODES * DIM;
  float* agg   = xr    + (size_t)NNODES * DIM;   // also h / centered h
  float* score = agg   + (size_t)NNODES * DIM;   // EP*4 (logit -> p)
  float* mmax  = score + (size_t)EPEDGE * NH;    // N*4
  float* denom = mmax  + (size_t)NNODES * NH;    // N*4
  float* gsum  = denom + (size_t)NNODES * NH;    // G*64
  float* vsum  = gsum  + (size_t)NG * DIM;       // G*64
  float* cnt   = vsum  + (size_t)NG * DIM;       // G

  const int elemBlocks = (NNODES * DIM + 255) / 256;           // 25000
  const int edgeBlocks = (EPEDGE + 7) / 8;                     // wave/edge, 8 waves/blk
  const int ehBlocks   = (EPEDGE * NH + 255) / 256;

  proj_wmma <<<NNODES / 32, 256, 0, stream>>>(x, Wl, Wr, xl, xr);
  init_ws   <<<elemBlocks, 256, 0, stream>>>(agg, mmax, denom, gsum, vsum, cnt);
  edge_score<<<edgeBlocks, 256, 0, stream>>>(xl, xr, ei, att, score, mmax);
  exp_denom <<<ehBlocks,   256, 0, stream>>>(ei, mmax, score, denom);
  aggregate <<<edgeBlocks, 256, 0, stream>>>(xl, ei, score, denom, agg);
  post_elu  <<<elemBlocks, 256, 0, stream>>>(x, bias_gat, batch, agg, gsum, cnt);
  center_var<<<elemBlocks, 256, 0, stream>>>(batch, gn_mscale, gsum, cnt, agg, vsum);
  finalize  <<<elemBlocks, 256, 0, stream>>>(batch, gn_weight, gn_bias, cnt, vsum, agg, out);
}